// InstanceDecoder_43817256353867
// MI455X (gfx1250) — compile-verified
//
#include <hip/hip_runtime.h>
#include <math.h>

typedef __attribute__((ext_vector_type(2))) float v2f;
typedef __attribute__((ext_vector_type(8))) float v8f;

#define N_HITS 16384
#define KDIM   256
#define DDIM   24
#define WPAD   32          // coord_w columns padded to 32 in LDS
#define BIGVAL 1e9f
#define STAGE_COLS 64      // columns staged into LDS per barrier round

// ---------------------------------------------------------------------------
// Kernel A: of = sigmoid(x @ beta_w + beta_b), one thread per hit row.
// ---------------------------------------------------------------------------
__global__ void beta_kernel(const float* __restrict__ x,
                            const float* __restrict__ bw,
                            const float* __restrict__ bb,
                            float* __restrict__ of) {
    int row = blockIdx.x * blockDim.x + threadIdx.x;
    const float4* xr = (const float4*)(x + (size_t)row * KDIM);
    const float4* w4 = (const float4*)bw;
    float acc = 0.0f;
#pragma unroll 4
    for (int i = 0; i < KDIM / 4; ++i) {
        float4 a = xr[i];
        float4 b = w4[i];
        acc += a.x * b.x + a.y * b.y + a.z * b.z + a.w * b.w;
    }
    acc += bb[0];
    of[row] = 1.0f / (1.0f + expf(-acc));
}

// ---------------------------------------------------------------------------
// Kernel B: ox = x @ coord_w + coord_b via V_WMMA_F32_16X16X4_F32.
// coord_w is staged zero-padded (24 -> 32 cols) into LDS once per block, so
// the K-loop has NO divergent loads: unconditional ds_load + global b64 + wmma.
// Also emits sq[m] = ||ox[m]||^2 via cross-lane shuffle reduction.
// ---------------------------------------------------------------------------
__global__ void coord_kernel(const float* __restrict__ x,
                             const float* __restrict__ cw,
                             const float* __restrict__ cb,
                             float* __restrict__ ox,
                             float* __restrict__ sq) {
    __shared__ float s_w[KDIM * WPAD];      // 32 KB, zero-padded weights

    int tid = threadIdx.x;
    // stage weights: thread k fills padded row k
    {
        const float* wr = cw + (size_t)tid * DDIM;
        float* dr = s_w + tid * WPAD;
#pragma unroll
        for (int j = 0; j < DDIM; ++j) dr[j] = wr[j];
#pragma unroll
        for (int j = DDIM; j < WPAD; ++j) dr[j] = 0.0f;
    }
    __syncthreads();

    int lane = tid & 31;
    int wid = blockIdx.x * (blockDim.x >> 5) + (tid >> 5);
    int rowBase = wid * 16;
    int lo = lane & 15;
    int hi = lane >> 4;
    int row = rowBase + lo;
    int n1 = 16 + lo;
    bool n1ok = (n1 < DDIM);

    v8f c0 = {};
    v8f c1 = {};
    const float* xr = x + (size_t)row * KDIM + 2 * hi;

    for (int kk = 0; kk < KDIM / 4; ++kk) {
        int k0 = 4 * kk;
        v2f a = *(const v2f*)(xr + k0);          // A[row][k0+2hi, k0+2hi+1]
        int krow = k0 + 2 * hi;
        v2f b0, b1;
        b0.x = s_w[krow * WPAD + lo];
        b0.y = s_w[(krow + 1) * WPAD + lo];
        b1.x = s_w[krow * WPAD + n1];
        b1.y = s_w[(krow + 1) * WPAD + n1];
        c0 = __builtin_amdgcn_wmma_f32_16x16x4_f32(false, a, false, b0,
                                                   (short)0, c0, false, false);
        c1 = __builtin_amdgcn_wmma_f32_16x16x4_f32(false, a, false, b1,
                                                   (short)0, c1, false, false);
    }

    float cb0 = cb[lo];
    float cb1 = n1ok ? cb[n1] : 0.0f;

    float vv0[8], vv1[8], sq8[8];
#pragma unroll
    for (int r = 0; r < 8; ++r) {
        vv0[r] = c0[r] + cb0;
        vv1[r] = c1[r] + cb1;                      // zero for n1 >= 24
        sq8[r] = vv0[r] * vv0[r] + (n1ok ? vv1[r] * vv1[r] : 0.0f);
    }
#pragma unroll
    for (int r = 0; r < 8; ++r) {
        int m = rowBase + 8 * hi + r;
        ox[(size_t)m * DDIM + lo] = vv0[r];
    }
    if (n1ok) {
#pragma unroll
        for (int r = 0; r < 8; ++r) {
            int m = rowBase + 8 * hi + r;
            ox[(size_t)m * DDIM + n1] = vv1[r];
        }
    }
    // reduce across the 16 lanes sharing each row (xor <= 8 stays in half-wave)
#pragma unroll
    for (int d = 1; d <= 8; d <<= 1) {
#pragma unroll
        for (int r = 0; r < 8; ++r) sq8[r] += __shfl_xor(sq8[r], d, 32);
    }
    if (lo == 0) {
#pragma unroll
        for (int r = 0; r < 8; ++r) sq[rowBase + 8 * hi + r] = sq8[r];
    }
}

// ---------------------------------------------------------------------------
// Kernel C: center_rank = cumsum(of > 0.1) - 1. One 1024-thread block,
// 16 elements per thread, Hillis-Steele scan of per-thread sums in LDS.
// ---------------------------------------------------------------------------
__global__ void rank_kernel(const float* __restrict__ of,
                            int* __restrict__ rank) {
    __shared__ int sums[1024];
    int tid = threadIdx.x;
    const int per = N_HITS / 1024;   // 16
    int base = tid * per;
    int local[per];
    int s = 0;
#pragma unroll
    for (int i = 0; i < per; ++i) {
        s += (of[base + i] > 0.1f) ? 1 : 0;
        local[i] = s;                 // inclusive within chunk
    }
    sums[tid] = s;
    __syncthreads();
    for (int off = 1; off < 1024; off <<= 1) {
        int v = (tid >= off) ? sums[tid - off] : 0;
        __syncthreads();
        sums[tid] += v;
        __syncthreads();
    }
    int excl = (tid == 0) ? 0 : sums[tid - 1];
#pragma unroll
    for (int i = 0; i < per; ++i) rank[base + i] = excl + local[i] - 1;
}

// ---------------------------------------------------------------------------
// Kernel D: fused pairwise-distance + masked argmin sweep.
// Block = 8 waves = 128 rows. All waves sweep the SAME column tiles:
// 64 columns of ox (+ norms + of) are staged into LDS cooperatively per
// barrier round, then each wave runs 4x {6 ds_load_b64 -> 6 WMMA -> min}.
// Global traffic cut 8x vs per-wave loads; inner loop is WMMA-dominated.
// ---------------------------------------------------------------------------
__global__ void assign_kernel(const float* __restrict__ ox,
                              const float* __restrict__ sq,
                              const float* __restrict__ of,
                              const int* __restrict__ rank,
                              float* __restrict__ minDistOut,
                              float* __restrict__ instOut) {
    __shared__ float s_ox[STAGE_COLS * DDIM];   // 6 KB
    __shared__ float s_sq[STAGE_COLS];
    __shared__ float s_of[STAGE_COLS];

    int tid = threadIdx.x;
    int lane = tid & 31;
    int lo = lane & 15;
    int hi = lane >> 4;
    int waveId = tid >> 5;
    int rowBase = blockIdx.x * 128 + waveId * 16;
    int row = rowBase + lo;

    // A fragments for this row block (K = 24 -> 6 chunks of 4)
    v2f aF[6];
#pragma unroll
    for (int kk = 0; kk < 6; ++kk)
        aF[kk] = *(const v2f*)(ox + (size_t)row * DDIM + 4 * kk + 2 * hi);

    float rowSq[8];
#pragma unroll
    for (int r = 0; r < 8; ++r) rowSq[r] = sq[rowBase + 8 * hi + r];

    float minV[8];
    int   minI[8];
#pragma unroll
    for (int r = 0; r < 8; ++r) { minV[r] = BIGVAL; minI[r] = 0; }

    for (int stage = 0; stage < N_HITS / STAGE_COLS; ++stage) {   // 256 rounds
        int colBase = stage * STAGE_COLS;
        const float* src = ox + (size_t)colBase * DDIM;

        __syncthreads();   // previous round's LDS reads complete
        // stage 64 cols * 24 floats = 1536 floats: 3 x b64 per thread
#pragma unroll
        for (int j = 0; j < 3; ++j) {
            int idx = (tid + j * 256) * 2;
            *(v2f*)&s_ox[idx] = *(const v2f*)&src[idx];
        }
        if (tid < STAGE_COLS) {
            s_sq[tid] = sq[colBase + tid];
            s_of[tid] = of[colBase + tid];
        }
        // speculative prefetch of the next stage (stays inside d_out)
        __builtin_prefetch(src + STAGE_COLS * DDIM + 6 * tid, 0, 0);
        __syncthreads();

#pragma unroll
        for (int tt = 0; tt < STAGE_COLS / 16; ++tt) {
            int c16 = tt * 16;
            const float* oc = &s_ox[(c16 + lo) * DDIM + 2 * hi];

            v8f c = {};
#pragma unroll
            for (int kk = 0; kk < 6; ++kk) {
                v2f b = *(const v2f*)(oc + 4 * kk);        // ds_load_b64
                c = __builtin_amdgcn_wmma_f32_16x16x4_f32(
                        false, aF[kk], false, b, (short)0, c, false, false);
            }

            float colSq = s_sq[c16 + lo];
            bool maskC = s_of[c16 + lo] > 0.1f;
            int col = colBase + c16 + lo;
#pragma unroll
            for (int r = 0; r < 8; ++r) {
                float d2 = rowSq[r] + colSq - 2.0f * c[r];
                float dist = sqrtf(fmaxf(d2, 0.0f));
                float val = (maskC && dist < 1.0f) ? dist : BIGVAL;
                if (val < minV[r]) { minV[r] = val; minI[r] = col; }
            }
        }
    }

    // cross-lane (val, idx) argmin reduction within each 16-lane half
#pragma unroll
    for (int d = 1; d <= 8; d <<= 1) {
#pragma unroll
        for (int r = 0; r < 8; ++r) {
            float ov = __shfl_xor(minV[r], d, 32);
            int   oi = __shfl_xor(minI[r], d, 32);
            if (ov < minV[r] || (ov == minV[r] && oi < minI[r])) {
                minV[r] = ov;
                minI[r] = oi;
            }
        }
    }

    if (lo == 0) {
#pragma unroll
        for (int r = 0; r < 8; ++r) {
            int m = rowBase + 8 * hi + r;
            float md = minV[r];
            minDistOut[m] = md;
            instOut[m] = (md < BIGVAL) ? (float)rank[minI[r]] : -1.0f;
        }
    }
}

// ---------------------------------------------------------------------------
extern "C" void kernel_launch(void* const* d_in, const int* in_sizes, int n_in,
                              void* d_out, int out_size, void* d_ws, size_t ws_size,
                              hipStream_t stream) {
    const float* x       = (const float*)d_in[0];
    const float* beta_w  = (const float*)d_in[1];
    const float* beta_b  = (const float*)d_in[2];
    const float* coord_w = (const float*)d_in[3];
    const float* coord_b = (const float*)d_in[4];

    // d_out layout: [ of (N) | ox (N*D) | min_dist (N) | instances (N) ]
    float* of      = (float*)d_out;
    float* oxOut   = of + N_HITS;
    float* minDist = oxOut + (size_t)N_HITS * DDIM;
    float* inst    = minDist + N_HITS;

    // workspace: sq (N floats) + rank (N ints)
    float* sq   = (float*)d_ws;
    int*   rank = (int*)(sq + N_HITS);

    // A: sigmoid matvec
    beta_kernel<<<N_HITS / 256, 256, 0, stream>>>(x, beta_w, beta_b, of);
    // B: coord GEMM via f32 WMMA (8 waves/block, 16 rows/wave)
    coord_kernel<<<N_HITS / (16 * 8), 256, 0, stream>>>(x, coord_w, coord_b,
                                                        oxOut, sq);
    // C: center rank prefix sum
    rank_kernel<<<1, 1024, 0, stream>>>(of, rank);
    // D: fused pairwise distance + masked argmin (8 waves share staged cols)
    assign_kernel<<<N_HITS / 128, 256, 0, stream>>>(oxOut, sq, of, rank,
                                                    minDist, inst);
}